// BinaryPathEncoder_14654428414481
// MI455X (gfx1250) — compile-verified
//
#include <hip/hip_runtime.h>

// BinaryPathEncoder on MI455X (gfx1250), compile-only tuned.
// Tree reformulation: table T[p] for p in [0, 2^17), built level-by-level as
// fp32 WMMA GEMMs (V_WMMA_F32_16X16X4_F32), then a bandwidth-bound gather.

typedef __attribute__((ext_vector_type(2))) float v2f;
typedef __attribute__((ext_vector_type(4))) float v4f;
typedef __attribute__((ext_vector_type(8))) float v8f;

#define DIM       256
#define TBL_ROWS  131072        // 2^17 tree nodes (row 0 == row 1 == identity)
#define MAX_LVL   16            // deepest tree level

// ---------------------------------------------------------------------------
// Transpose primitives: Pt[c][k][n] = P[c][n][k]  (512 KB once, stays in L2)
// ---------------------------------------------------------------------------
__global__ void k_transpose(const float* __restrict__ P, float* __restrict__ Pt) {
    int c = blockIdx.y;           // primitive 0/1
    int k = blockIdx.x;           // source column
    int n = threadIdx.x;          // source row
    Pt[(size_t)c * DIM * DIM + (size_t)k * DIM + n] =
        P[(size_t)c * DIM * DIM + (size_t)n * DIM + k];
}

// ---------------------------------------------------------------------------
// Seed table rows 0 and 1 with the identity row vector.
// ---------------------------------------------------------------------------
__global__ void k_init(const float* __restrict__ identity, float* __restrict__ tbl) {
    int k = threadIdx.x;
    float v = identity[k];
    tbl[k]       = v;   // p = 0 (never descends)
    tbl[DIM + k] = v;   // p = 1 (depth-0 node)
}

// ---------------------------------------------------------------------------
// Tiny levels d = 1..4 (2..16 child rows): plain VALU matvec.
// child j (blockIdx.x) at table row 2^d + j; parent = 2^(d-1) + (j mod 2^(d-1));
// primitive c = j >> (d-1).
// ---------------------------------------------------------------------------
__global__ void k_level_small(const float* __restrict__ P, float* __restrict__ tbl, int d) {
    int j    = blockIdx.x;
    int half = 1 << (d - 1);
    int c    = j >> (d - 1);
    int jj   = j & (half - 1);
    const float* parent = tbl + (size_t)(half + jj) * DIM;
    const float* Prow   = P + (size_t)c * DIM * DIM + (size_t)threadIdx.x * DIM;
    float s = 0.f;
    #pragma unroll 8
    for (int k = 0; k < DIM; ++k) s += parent[k] * Prow[k];
    tbl[(size_t)((1 << d) + j) * DIM + threadIdx.x] = s;
}

// ---------------------------------------------------------------------------
// WMMA levels d = 5..16: children[c*half + m, :] = Parents[m, :] @ P_c^T.
// Grid: (half/16, 2), 256 threads = 8 wave32. Block stages a 16x256 fp32
// A-tile in LDS (row stride 260 floats: banks spread, 16B-aligned float4
// stores, 8B-aligned float2 ds loads). Wave w computes column tiles
// [32w,32w+16) and [32w+16,32w+32); one A fragment feeds both WMMAs.
// ---------------------------------------------------------------------------
__global__ __launch_bounds__(256, 2)
void k_level_wmma(const float* __restrict__ tbl_parents,
                  float* __restrict__ tbl_children,
                  const float* __restrict__ Pt, int half) {
    __shared__ float As[16][260];

    const int mtile = blockIdx.x;
    const int c     = blockIdx.y;
    const float* parents = tbl_parents + (size_t)mtile * 16 * DIM;

    // Cooperative A-tile load: 16 rows x 64 float4 = 1024 float4 / 256 threads.
    const int t = threadIdx.x;
    #pragma unroll
    for (int i = 0; i < 4; ++i) {
        int idx  = t + i * 256;      // float4 slot id
        int r    = idx >> 6;         // row (64 float4 per row)
        int col4 = idx & 63;
        v4f val = *reinterpret_cast<const v4f*>(parents + (size_t)r * DIM + col4 * 4);
        *reinterpret_cast<v4f*>(&As[r][col4 * 4]) = val;
    }
    __syncthreads();

    const int lane = t & 31;
    const int wave = t >> 5;
    const int m    = lane & 15;   // A row within tile
    const int h    = lane >> 4;   // half-wave selector (K pairs 0/1 vs 2/3)
    const int ncol = wave * 32 + m;

    const float* Ptc = Pt + (size_t)c * DIM * DIM;
    const float* B0  = Ptc + ncol;        // tile 0 column
    const float* B1  = Ptc + ncol + 16;   // tile 1 column

    v8f acc0 = {0, 0, 0, 0, 0, 0, 0, 0};
    v8f acc1 = {0, 0, 0, 0, 0, 0, 0, 0};

    #pragma unroll 4
    for (int k0 = 0; k0 < DIM; k0 += 4) {
        int kk = k0 + 2 * h;
        // A fragment: 16x4 fp32; lane holds A[m][kk], A[m][kk+1]  (ds_load_b64)
        v2f a = *reinterpret_cast<const v2f*>(&As[m][kk]);
        // B fragments: 4x16 fp32; lane holds B[kk][n], B[kk+1][n]
        v2f b0; b0.x = B0[(size_t)kk * DIM]; b0.y = B0[(size_t)(kk + 1) * DIM];
        v2f b1; b1.x = B1[(size_t)kk * DIM]; b1.y = B1[(size_t)(kk + 1) * DIM];
        acc0 = __builtin_amdgcn_wmma_f32_16x16x4_f32(false, a, false, b0,
                                                     (short)0, acc0, false, false);
        acc1 = __builtin_amdgcn_wmma_f32_16x16x4_f32(false, a, false, b1,
                                                     (short)0, acc1, false, false);
    }

    // C/D layout: VGPR i -> row i (lanes 0-15) or row i+8 (lanes 16-31), col = lane&15.
    float* outp = tbl_children + ((size_t)c * half + (size_t)mtile * 16) * DIM;
    const int col = lane & 15;
    #pragma unroll
    for (int i = 0; i < 8; ++i) {
        int row = i + 8 * h;
        outp[(size_t)row * DIM + wave * 32 + col]      = acc0[i];
        outp[(size_t)row * DIM + wave * 32 + 16 + col] = acc1[i];
    }
}

// ---------------------------------------------------------------------------
// Gather: out[n] = T[unique[n]], float4 granularity (pure bandwidth).
// ---------------------------------------------------------------------------
__global__ void k_gather(const int* __restrict__ unique, const float* __restrict__ tbl,
                         float* __restrict__ out, int n) {
    int tid = blockIdx.x * blockDim.x + threadIdx.x;   // over n * 64 float4
    int row = tid >> 6;
    int q   = tid & 63;
    if (row < n) {
        int p = unique[row];
        const v4f* src = reinterpret_cast<const v4f*>(tbl + (size_t)p * DIM);
        v4f*       dst = reinterpret_cast<v4f*>(out + (size_t)row * DIM);
        dst[q] = src[q];
    }
}

// ---------------------------------------------------------------------------
// Fallback (workspace too small): direct per-row chain, LDS ping-pong matvec.
// ---------------------------------------------------------------------------
__global__ void k_direct(const int* __restrict__ unique, const float* __restrict__ P,
                         const float* __restrict__ identity, float* __restrict__ out) {
    __shared__ float buf[2][DIM];
    const int n = blockIdx.x, t = threadIdx.x;
    const int p = unique[n];
    buf[0][t] = identity[t];
    __syncthreads();
    int cur = 0;
    for (int d = 0; d < 17; ++d) {
        if (p >= (1 << (d + 1))) {              // uniform per block
            int c = (p >> d) & 1;
            const float* Prow = P + (size_t)c * DIM * DIM + (size_t)t * DIM;
            float s = 0.f;
            #pragma unroll 8
            for (int k = 0; k < DIM; ++k) s += buf[cur][k] * Prow[k];
            buf[1 - cur][t] = s;
            cur = 1 - cur;
            __syncthreads();
        }
    }
    out[(size_t)n * DIM + t] = buf[cur][t];
}

// ---------------------------------------------------------------------------
extern "C" void kernel_launch(void* const* d_in, const int* in_sizes, int n_in,
                              void* d_out, int out_size, void* d_ws, size_t ws_size,
                              hipStream_t stream) {
    const int*   unique   = (const int*)d_in[0];
    const float* prim     = (const float*)d_in[1];
    const float* identity = (const float*)d_in[2];
    float*       out      = (float*)d_out;
    const int    n        = in_sizes[0];

    const size_t tbl_bytes = (size_t)TBL_ROWS * DIM * sizeof(float);
    const size_t pt_bytes  = (size_t)2 * DIM * DIM * sizeof(float);

    if (ws_size >= tbl_bytes + pt_bytes) {
        float* tbl = (float*)d_ws;
        float* Pt  = tbl + (size_t)TBL_ROWS * DIM;

        k_transpose<<<dim3(DIM, 2), DIM, 0, stream>>>(prim, Pt);
        k_init<<<1, DIM, 0, stream>>>(identity, tbl);

        for (int d = 1; d <= 4; ++d)
            k_level_small<<<(1 << d), DIM, 0, stream>>>(prim, tbl, d);

        for (int d = 5; d <= MAX_LVL; ++d) {
            int half = 1 << (d - 1);
            k_level_wmma<<<dim3(half / 16, 2), 256, 0, stream>>>(
                tbl + (size_t)half * DIM,            // parents base
                tbl + ((size_t)1 << d) * DIM,        // children base
                Pt, half);
        }

        int total = n * 64;  // float4 elements
        k_gather<<<(total + 255) / 256, 256, 0, stream>>>(unique, tbl, out, n);
    } else {
        k_direct<<<n, DIM, 0, stream>>>(unique, prim, identity, out);
    }
}